// GalerkinTransformerDecoderLayer_47244640256555
// MI455X (gfx1250) — compile-verified
//
#include <hip/hip_runtime.h>
#include <stdint.h>

typedef float v2f __attribute__((ext_vector_type(2)));
typedef float v8f __attribute__((ext_vector_type(8)));

#define BN 2
#define NN 4096
#define DD 256
#define HH 8
#define DKK 32
#define DFF_ 1024
#define EPS_LN_ 1e-5f
#define EPS_CAUSAL_ 1e-7f
#define SCAN_C 64
#define SCAN_L 64
#define KTV_C 32
#define KTV_L 128

__device__ __forceinline__ float wave_sum32(float x) {
#pragma unroll
  for (int off = 16; off > 0; off >>= 1) x += __shfl_xor(x, off, 32);
  return x;
}

__device__ __forceinline__ v8f wmma4(v2f a, v2f b, v8f c) {
  return __builtin_amdgcn_wmma_f32_16x16x4_f32(false, a, false, b, (short)0, c,
                                               false, false);
}

// ---------------------------------------------------------------------------
// GEMM: C[M,Nc] = A[M,K] @ W[K,Nc] + bias, optional ReLU.
// One wave computes a 32x32 output tile as 2x2 WMMA tiles (fragment reuse:
// 4 v_wmma_f32_16x16x4_f32 per K-step for 2 A-frag + 2 B-frag loads).
// Bias staged through LDS with the CDNA5 async-load path (ASYNCcnt).
// ---------------------------------------------------------------------------
__global__ __launch_bounds__(128) void gemm_bias_kernel(
    const float* __restrict__ A, const float* __restrict__ W,
    const float* __restrict__ bias, float* __restrict__ C, int M, int K,
    int Nc, int relu) {
  __shared__ float sbias[4 * 32];
  const int lane = threadIdx.x & 31;
  const int wave = threadIdx.x >> 5;
  const int tile = blockIdx.x * 4 + wave;
  const int ntn = Nc >> 5;               // 32-wide column tiles
  const int m0 = (tile / ntn) << 5;
  const int n0 = (tile % ntn) << 5;
  if (m0 >= M) return;
  const int half = lane >> 4;  // 0: K rows k0..k0+1, 1: K rows k0+2..k0+3
  const int l15 = lane & 15;

  // CDNA5 async load of the 32-wide bias tile into LDS (tracked by ASYNCcnt).
  {
    unsigned lds_off = (unsigned)(size_t)&sbias[wave * 32 + lane];
    unsigned long long gp = (unsigned long long)(bias + n0 + lane);
    asm volatile("global_load_async_to_lds_b32 %0, %1, off" ::"v"(lds_off),
                 "v"(gp)
                 : "memory");
  }

  v8f c00 = {}, c01 = {}, c10 = {}, c11 = {};
  const float* ap0 = A + (size_t)(m0 + l15) * K + half * 2;
  const float* ap1 = A + (size_t)(m0 + 16 + l15) * K + half * 2;
  const float* wp0 = W + (size_t)(half * 2) * Nc + n0 + l15;
  const float* wp1 = W + (size_t)(half * 2) * Nc + n0 + 16 + l15;
#pragma unroll 4
  for (int k0 = 0; k0 < K; k0 += 4) {
    v2f a0, a1, b0, b1;
    a0.x = ap0[k0];
    a0.y = ap0[k0 + 1];
    a1.x = ap1[k0];
    a1.y = ap1[k0 + 1];
    b0.x = wp0[(size_t)k0 * Nc];
    b0.y = wp0[(size_t)(k0 + 1) * Nc];
    b1.x = wp1[(size_t)k0 * Nc];
    b1.y = wp1[(size_t)(k0 + 1) * Nc];
    c00 = wmma4(a0, b0, c00);
    c01 = wmma4(a0, b1, c01);
    c10 = wmma4(a1, b0, c10);
    c11 = wmma4(a1, b1, c11);
  }

  asm volatile("s_wait_asynccnt 0" ::: "memory");
  float bc0 = sbias[wave * 32 + l15];
  float bc1 = sbias[wave * 32 + 16 + l15];
#pragma unroll
  for (int i = 0; i < 8; ++i) {
    int r = m0 + i + half * 8;  // C layout: VGPR i -> M=i / M=i+8
    float v00 = c00[i] + bc0;
    float v01 = c01[i] + bc1;
    float v10 = c10[i] + bc0;
    float v11 = c11[i] + bc1;
    if (relu) {
      v00 = fmaxf(v00, 0.0f);
      v01 = fmaxf(v01, 0.0f);
      v10 = fmaxf(v10, 0.0f);
      v11 = fmaxf(v11, 0.0f);
    }
    C[(size_t)r * Nc + n0 + l15] = v00;
    C[(size_t)r * Nc + n0 + 16 + l15] = v01;
    C[(size_t)(r + 16) * Nc + n0 + l15] = v10;
    C[(size_t)(r + 16) * Nc + n0 + 16 + l15] = v11;
  }
}

// ---------------------------------------------------------------------------
// Per-head LayerNorm over DK=32 (one wave per (b,n,h)); optional post-scale.
// ---------------------------------------------------------------------------
__global__ __launch_bounds__(256) void head_ln_kernel(
    float* __restrict__ Y, const float* __restrict__ g,
    const float* __restrict__ beta, float scale) {
  int lane = threadIdx.x & 31;
  int wave = threadIdx.x >> 5;
  int gid = blockIdx.x * 8 + wave;  // flat (b*N+n)*H + h
  int h = gid % HH;
  int bn = gid / HH;
  size_t idx = (size_t)bn * DD + h * DKK + lane;
  float xv = Y[idx];
  float m = wave_sum32(xv) * (1.0f / DKK);
  float d = xv - m;
  float var = wave_sum32(d * d) * (1.0f / DKK);
  float r = rsqrtf(var + EPS_LN_);
  Y[idx] = (d * r * g[h * DKK + lane] + beta[h * DKK + lane]) * scale;
}

__global__ void zero_kernel(float* __restrict__ p, int n) {
  int i = blockIdx.x * blockDim.x + threadIdx.x;
  if (i < n) p[i] = 0.0f;
}

// ---------------------------------------------------------------------------
// ktv[b,h] += K^T @ V over an N-chunk (WMMA, A = K columns), atomic reduce.
// ---------------------------------------------------------------------------
__global__ __launch_bounds__(128) void ktv_accum_kernel(
    const float* __restrict__ Kmat, const float* __restrict__ Vmat,
    float* __restrict__ ktv) {
  int lane = threadIdx.x & 31;
  int wave = threadIdx.x >> 5;
  int wid = blockIdx.x * 4 + wave;
  int c = wid % KTV_C;
  int bh = wid / KTV_C;
  int h = bh % HH;
  int b = bh / HH;
  int half = lane >> 4;
  int l15 = lane & 15;
  v8f a00 = {}, a01 = {}, a10 = {}, a11 = {};
  int n0 = c * KTV_L;
  for (int n = n0; n < n0 + KTV_L; n += 4) {
    size_t base = ((size_t)b * NN + n + half * 2) * DD + h * DKK;
    const float* kp = Kmat + base;
    const float* vp = Vmat + base;
    v2f fa0, fa1, fb0, fb1;
    fa0.x = kp[l15];
    fa0.y = kp[DD + l15];
    fa1.x = kp[16 + l15];
    fa1.y = kp[DD + 16 + l15];
    fb0.x = vp[l15];
    fb0.y = vp[DD + l15];
    fb1.x = vp[16 + l15];
    fb1.y = vp[DD + 16 + l15];
    a00 = wmma4(fa0, fb0, a00);
    a01 = wmma4(fa0, fb1, a01);
    a10 = wmma4(fa1, fb0, a10);
    a11 = wmma4(fa1, fb1, a11);
  }
  float* out = ktv + (size_t)bh * DKK * DKK;
#pragma unroll
  for (int i = 0; i < 8; ++i) {
    int r0 = i + half * 8;
    atomicAdd(out + r0 * DKK + l15, a00[i]);
    atomicAdd(out + r0 * DKK + 16 + l15, a01[i]);
    atomicAdd(out + (16 + r0) * DKK + l15, a10[i]);
    atomicAdd(out + (16 + r0) * DKK + 16 + l15, a11[i]);
  }
}

// ---------------------------------------------------------------------------
// Out[b,n,h,:] = Q[b,n,h,:] @ (ktv[b,h]*invN)   (WMMA, K=32, 16x32 per wave,
// A fragment reused across both 16-wide e-tiles)
// ---------------------------------------------------------------------------
__global__ __launch_bounds__(128) void galerkin_out_kernel(
    const float* __restrict__ Q, const float* __restrict__ ktv,
    float* __restrict__ Out, float invN) {
  int lane = threadIdx.x & 31;
  int wave = threadIdx.x >> 5;
  int wid = blockIdx.x * 4 + wave;
  int mt = wid % (NN / 16);
  wid /= (NN / 16);
  int h = wid % HH;
  int b = wid / HH;
  int m0 = mt * 16;
  int half = lane >> 4, l15 = lane & 15;
  const float* qp = Q + ((size_t)b * NN + m0 + l15) * DD + h * DKK + half * 2;
  const float* tp0 =
      ktv + (size_t)(b * HH + h) * DKK * DKK + (half * 2) * DKK + l15;
  const float* tp1 = tp0 + 16;
  v8f acc0 = {}, acc1 = {};
#pragma unroll
  for (int k0 = 0; k0 < DKK; k0 += 4) {
    v2f a, b0, b1;
    a.x = qp[k0];
    a.y = qp[k0 + 1];
    b0.x = tp0[k0 * DKK] * invN;
    b0.y = tp0[(k0 + 1) * DKK] * invN;
    b1.x = tp1[k0 * DKK] * invN;
    b1.y = tp1[(k0 + 1) * DKK] * invN;
    acc0 = wmma4(a, b0, acc0);
    acc1 = wmma4(a, b1, acc1);
  }
#pragma unroll
  for (int i = 0; i < 8; ++i) {
    int row = m0 + i + half * 8;
    size_t ob = ((size_t)b * NN + row) * DD + h * DKK;
    Out[ob + l15] = acc0[i];
    Out[ob + 16 + l15] = acc1[i];
  }
}

// ---------------------------------------------------------------------------
// Out = LN(A + R) * g + beta over D=256 (one wave per row).
// ---------------------------------------------------------------------------
__global__ __launch_bounds__(256) void add_ln_kernel(
    const float* __restrict__ A, const float* __restrict__ R,
    const float* __restrict__ g, const float* __restrict__ beta,
    float* __restrict__ Out) {
  int lane = threadIdx.x & 31;
  int wave = threadIdx.x >> 5;
  int row = blockIdx.x * 8 + wave;
  size_t base = (size_t)row * DD;
  float v[8];
  float s = 0.0f;
#pragma unroll
  for (int i = 0; i < 8; ++i) {
    v[i] = A[base + lane + i * 32] + R[base + lane + i * 32];
    s += v[i];
  }
  float mean = wave_sum32(s) * (1.0f / DD);
  float ss = 0.0f;
#pragma unroll
  for (int i = 0; i < 8; ++i) {
    float d = v[i] - mean;
    ss += d * d;
  }
  float var = wave_sum32(ss) * (1.0f / DD);
  float r = rsqrtf(var + EPS_LN_);
#pragma unroll
  for (int i = 0; i < 8; ++i) {
    int col = lane + i * 32;
    Out[base + col] = (v[i] - mean) * r * g[col] + beta[col];
  }
}

// ---------------------------------------------------------------------------
// Causal linear attention: chunked prefix scan over N.
// State per (b,h): kcum[32] (lane d) + kv[32][32] (lane e holds column e).
// ---------------------------------------------------------------------------
__global__ __launch_bounds__(256) void scan_pass1_kernel(
    const float* __restrict__ Kmat, const float* __restrict__ Vmat,
    const unsigned char* __restrict__ mask, float* __restrict__ Sk,
    float* __restrict__ Skv) {
  int lane = threadIdx.x & 31;
  int wave = threadIdx.x >> 5;
  int wid = blockIdx.x * 8 + wave;
  int c = wid % SCAN_C;
  int bh = wid / SCAN_C;
  int h = bh % HH, b = bh / HH;
  float sk = 0.0f;
  float skv[DKK];
#pragma unroll
  for (int d = 0; d < DKK; ++d) skv[d] = 0.0f;
  int n0 = c * SCAN_L;
  for (int n = n0; n < n0 + SCAN_L; ++n) {
    float mf = mask[b * NN + n] ? 1.0f : 0.0f;
    size_t base = ((size_t)b * NN + n) * DD + h * DKK;
    float kl = Kmat[base + lane] * mf;
    float vl = Vmat[base + lane] * mf;
    sk += kl;
#pragma unroll
    for (int d = 0; d < DKK; ++d) skv[d] += __shfl(kl, d, 32) * vl;
  }
  size_t cb = (size_t)bh * SCAN_C + c;
  Sk[cb * DKK + lane] = sk;
#pragma unroll
  for (int d = 0; d < DKK; ++d) Skv[(cb * DKK + d) * DKK + lane] = skv[d];
}

__global__ __launch_bounds__(256) void scan_pass2_kernel(
    const float* __restrict__ Sk, const float* __restrict__ Skv,
    float* __restrict__ Pk, float* __restrict__ Pkv) {
  int bh = blockIdx.x;
  int t = threadIdx.x;
  float runk = 0.0f;
  float runv[4] = {0.0f, 0.0f, 0.0f, 0.0f};
  for (int c = 0; c < SCAN_C; ++c) {
    size_t cb = (size_t)bh * SCAN_C + c;
    if (t < DKK) {
      size_t i = cb * DKK + t;
      Pk[i] = runk;
      runk += Sk[i];
    }
    size_t vb = cb * DKK * DKK + t * 4;
#pragma unroll
    for (int j = 0; j < 4; ++j) {
      Pkv[vb + j] = runv[j];
      runv[j] += Skv[vb + j];
    }
  }
}

__global__ __launch_bounds__(256) void scan_pass3_kernel(
    const float* __restrict__ Q, const float* __restrict__ Kmat,
    const float* __restrict__ Vmat, const unsigned char* __restrict__ mask,
    const float* __restrict__ Pk, const float* __restrict__ Pkv,
    float* __restrict__ Out) {
  int lane = threadIdx.x & 31;
  int wave = threadIdx.x >> 5;
  int wid = blockIdx.x * 8 + wave;
  int c = wid % SCAN_C;
  int bh = wid / SCAN_C;
  int h = bh % HH, b = bh / HH;
  size_t cb = (size_t)bh * SCAN_C + c;
  float kc = Pk[cb * DKK + lane];
  float kv[DKK];
#pragma unroll
  for (int d = 0; d < DKK; ++d) kv[d] = Pkv[(cb * DKK + d) * DKK + lane];
  int n0 = c * SCAN_L;
  for (int n = n0; n < n0 + SCAN_L; ++n) {
    float mf = mask[b * NN + n] ? 1.0f : 0.0f;
    size_t base = ((size_t)b * NN + n) * DD + h * DKK;
    float kl = Kmat[base + lane] * mf;
    float vl = Vmat[base + lane] * mf;
    float ql = Q[base + lane];
    kc += kl;
#pragma unroll
    for (int d = 0; d < DKK; ++d) kv[d] += __shfl(kl, d, 32) * vl;
    float denom = wave_sum32((kc + EPS_CAUSAL_) * ql);
    float o = 0.0f;
#pragma unroll
    for (int d = 0; d < DKK; ++d) o += __shfl(ql, d, 32) * kv[d];
    Out[base + lane] = o / denom;
  }
}

// ---------------------------------------------------------------------------
extern "C" void kernel_launch(void* const* d_in, const int* in_sizes, int n_in,
                              void* d_out, int out_size, void* d_ws,
                              size_t ws_size, hipStream_t stream) {
  const float* x = (const float*)d_in[0];
  const float* memi = (const float*)d_in[1];
  const unsigned char* mask = (const unsigned char*)d_in[2];
  const float* sa_q_w = (const float*)d_in[3];
  const float* sa_q_b = (const float*)d_in[4];
  const float* sa_k_w = (const float*)d_in[5];
  const float* sa_k_b = (const float*)d_in[6];
  const float* sa_v_w = (const float*)d_in[7];
  const float* sa_v_b = (const float*)d_in[8];
  const float* ca_q_w = (const float*)d_in[9];
  const float* ca_q_b = (const float*)d_in[10];
  const float* ca_k_w = (const float*)d_in[11];
  const float* ca_k_b = (const float*)d_in[12];
  const float* ca_v_w = (const float*)d_in[13];
  const float* ca_v_b = (const float*)d_in[14];
  const float* ff1_w = (const float*)d_in[15];
  const float* ff1_b = (const float*)d_in[16];
  const float* ff2_w = (const float*)d_in[17];
  const float* ff2_b = (const float*)d_in[18];
  const float* sa_k_ln_g = (const float*)d_in[19];
  const float* sa_k_ln_b = (const float*)d_in[20];
  const float* sa_v_ln_g = (const float*)d_in[21];
  const float* sa_v_ln_b = (const float*)d_in[22];
  const float* ca_k_ln_g = (const float*)d_in[23];
  const float* ca_k_ln_b = (const float*)d_in[24];
  const float* ca_q_ln_g = (const float*)d_in[25];
  const float* ca_q_ln_b = (const float*)d_in[26];
  const float* ln1_g = (const float*)d_in[27];
  const float* ln1_b = (const float*)d_in[28];
  const float* ln2_g = (const float*)d_in[29];
  const float* ln2_b = (const float*)d_in[30];
  const float* ln3_g = (const float*)d_in[31];
  const float* ln3_b = (const float*)d_in[32];

  float* ws = (float*)d_ws;
  const size_t SZ = (size_t)BN * NN * DD;  // 2,097,152
  float* qb = ws;
  float* kb = ws + SZ;
  float* vb = ws + 2 * SZ;
  float* x1 = ws + 3 * SZ;
  float* attn = ws + 4 * SZ;
  float* x2 = ws + 5 * SZ;
  float* hb = ws + 6 * SZ;                        // B*N*DFF = 8,388,608
  float* ktv = hb + (size_t)BN * NN * DFF_;       // 16,384
  float* Sk = ktv + (size_t)BN * HH * DKK * DKK;  // 32,768
  float* Skv = Sk + (size_t)BN * HH * SCAN_C * DKK;        // 1,048,576
  float* Pk = Skv + (size_t)BN * HH * SCAN_C * DKK * DKK;  // 32,768
  float* Pkv = Pk + (size_t)BN * HH * SCAN_C * DKK;        // 1,048,576
  float* out = (float*)d_out;

  const int M = BN * NN;  // 8192
  const float invN = 1.0f / (float)NN;

  // 32x32 tiles per wave, 4 waves per block.
  const int g_proj = (M / 32) * (DD / 32) / 4;   // 512
  const int g_ff1 = (M / 32) * (DFF_ / 32) / 4;  // 2048
  const int g_ff2 = (M / 32) * (DD / 32) / 4;    // 512

  // ---- Galerkin self-attention -------------------------------------------
  gemm_bias_kernel<<<g_proj, 128, 0, stream>>>(x, sa_q_w, sa_q_b, qb, M, DD, DD, 0);
  gemm_bias_kernel<<<g_proj, 128, 0, stream>>>(x, sa_k_w, sa_k_b, kb, M, DD, DD, 0);
  gemm_bias_kernel<<<g_proj, 128, 0, stream>>>(x, sa_v_w, sa_v_b, vb, M, DD, DD, 0);
  head_ln_kernel<<<8192, 256, 0, stream>>>(kb, sa_k_ln_g, sa_k_ln_b, 1.0f);
  head_ln_kernel<<<8192, 256, 0, stream>>>(vb, sa_v_ln_g, sa_v_ln_b, 1.0f);
  zero_kernel<<<64, 256, 0, stream>>>(ktv, BN * HH * DKK * DKK);
  ktv_accum_kernel<<<128, 128, 0, stream>>>(kb, vb, ktv);
  galerkin_out_kernel<<<2048, 128, 0, stream>>>(qb, ktv, attn, invN);
  add_ln_kernel<<<1024, 256, 0, stream>>>(x, attn, ln1_g, ln1_b, x1);

  // ---- Causal (linear) cross-attention -----------------------------------
  gemm_bias_kernel<<<g_proj, 128, 0, stream>>>(x1, ca_q_w, ca_q_b, qb, M, DD, DD, 0);
  gemm_bias_kernel<<<g_proj, 128, 0, stream>>>(memi, ca_k_w, ca_k_b, kb, M, DD, DD, 0);
  gemm_bias_kernel<<<g_proj, 128, 0, stream>>>(memi, ca_v_w, ca_v_b, vb, M, DD, DD, 0);
  head_ln_kernel<<<8192, 256, 0, stream>>>(qb, ca_q_ln_g, ca_q_ln_b, 1.0f);
  head_ln_kernel<<<8192, 256, 0, stream>>>(kb, ca_k_ln_g, ca_k_ln_b, invN);
  scan_pass1_kernel<<<128, 256, 0, stream>>>(kb, vb, mask, Sk, Skv);
  scan_pass2_kernel<<<BN * HH, 256, 0, stream>>>(Sk, Skv, Pk, Pkv);
  scan_pass3_kernel<<<128, 256, 0, stream>>>(qb, kb, vb, mask, Pk, Pkv, attn);
  add_ln_kernel<<<1024, 256, 0, stream>>>(x1, attn, ln2_g, ln2_b, x2);

  // ---- FFN ---------------------------------------------------------------
  gemm_bias_kernel<<<g_ff1, 128, 0, stream>>>(x2, ff1_w, ff1_b, hb, M, DD, DFF_, 1);
  gemm_bias_kernel<<<g_ff2, 128, 0, stream>>>(hb, ff2_w, ff2_b, qb, M, DFF_, DD, 0);
  add_ln_kernel<<<1024, 256, 0, stream>>>(x2, qb, ln3_g, ln3_b, out);
}